// MULTModel_9663676416806
// MI455X (gfx1250) — compile-verified
//
#include <hip/hip_runtime.h>
#include <math.h>

// ---- model dims ----
#define Bz   8
#define Tz   150
#define Dz   30
#define Hz   5
#define HDz  6
#define Lz   3
#define BHz  40          // B*H
#define TP   160         // padded T (10 tiles of 16)
#define HDP  8           // padded head dim (2 k-chunks of 4)
#define ROWS 1200        // T*B
#define NTBD 36000       // T*B*D

typedef __attribute__((ext_vector_type(2))) float v2f;
typedef __attribute__((ext_vector_type(8))) float v8f;
typedef int v4i __attribute__((vector_size(4 * sizeof(int))));

#if __has_builtin(__builtin_amdgcn_global_load_async_to_lds_b128)
#define ASYNC_LDS 1
#endif

// ---------------- input modality projections: [B,T,K] @ W[D,K]^T -> [T,B,D]
__global__ void proj_kernel(const float* __restrict__ x, const float* __restrict__ W,
                            float* __restrict__ out, int Kdim) {
  int idx = blockIdx.x * blockDim.x + threadIdx.x;
  if (idx >= NTBD) return;
  int d = idx % Dz;
  int b = (idx / Dz) % Bz;
  int t = idx / (Dz * Bz);
  const float* xr = x + (b * Tz + t) * Kdim;
  const float* wr = W + d * Kdim;
  float s = 0.f;
  for (int k = 0; k < Kdim; ++k) s += xr[k] * wr[k];
  out[idx] = s;
}

// ---------------- layernorm over last dim (biased var), rows = T*B
__global__ void ln_kernel(const float* __restrict__ x, const float* __restrict__ g,
                          const float* __restrict__ bb, float* __restrict__ y) {
  int r = blockIdx.x * blockDim.x + threadIdx.x;
  if (r >= ROWS) return;
  const float* xr = x + r * Dz;
  float m = 0.f;
  for (int j = 0; j < Dz; ++j) m += xr[j];
  m *= (1.0f / Dz);
  float v = 0.f;
  for (int j = 0; j < Dz; ++j) { float t0 = xr[j] - m; v += t0 * t0; }
  v *= (1.0f / Dz);
  float inv = rsqrtf(v + 1e-5f);
  float* yr = y + r * Dz;
  for (int j = 0; j < Dz; ++j) yr[j] = (xr[j] - m) * inv * g[j] + bb[j];
}

// ---------------- q/k/v projections (q pre-scaled by HD^-0.5)
__global__ void qkv_kernel(const float* __restrict__ xn, const float* __restrict__ kn,
                           const float* __restrict__ vn, const float* __restrict__ iw,
                           const float* __restrict__ ib, float* __restrict__ q,
                           float* __restrict__ k, float* __restrict__ v, float scale) {
  int idx = blockIdx.x * blockDim.x + threadIdx.x;
  if (idx >= ROWS * Dz) return;
  int d = idx % Dz;
  int r = idx / Dz;
  const float* wq = iw + d * Dz;
  const float* wk = iw + (Dz + d) * Dz;
  const float* wv = iw + (2 * Dz + d) * Dz;
  const float* xr = xn + r * Dz;
  const float* kr = kn + r * Dz;
  const float* vr = vn + r * Dz;
  float sq = 0.f, sk = 0.f, sv = 0.f;
  for (int e = 0; e < Dz; ++e) { sq += xr[e] * wq[e]; sk += kr[e] * wk[e]; sv += vr[e] * wv[e]; }
  q[idx] = (sq + ib[d]) * scale;
  k[idx] = sk + ib[Dz + d];
  v[idx] = sv + ib[2 * Dz + d];
}

// ---------------- split heads + zero-pad: [T,B,D] -> [BH][TP][HDP]
__global__ void heads_kernel(const float* __restrict__ q, const float* __restrict__ k,
                             const float* __restrict__ v, float* __restrict__ qh,
                             float* __restrict__ kh, float* __restrict__ vh) {
  int idx = blockIdx.x * blockDim.x + threadIdx.x;
  if (idx >= BHz * TP * HDP) return;
  int e = idx % HDP;
  int t = (idx / HDP) % TP;
  int i = idx / (HDP * TP);
  int b = i / Hz, hh = i % Hz;
  float fq = 0.f, fk = 0.f, fv = 0.f;
  if (t < Tz && e < HDz) {
    int src = (t * Bz + b) * Dz + hh * HDz + e;
    fq = q[src]; fk = k[src]; fv = v[src];
  }
  qh[idx] = fq; kh[idx] = fk; vh[idx] = fv;
}

// ---------------- WMMA trilinear scores + fused mean/max over c
// One wave per (i, a): S[b,c] = sum_t K[b,t] * (q_a[t] * V[c,t]); write
// fused[i][a][b] = mean_c S + max_c S  for b < Tz.
// K tile staged to LDS (async DMA if available); W[c][t]=q_a[t]*V[c][t]
// precomputed once into LDS so the 200-WMMA inner loop reads only LDS.
__global__ __launch_bounds__(32)
void attn_scores_kernel(const float* __restrict__ qh, const float* __restrict__ kh,
                        const float* __restrict__ vh, float* __restrict__ fused) {
  __shared__ float ldsK[TP * HDP];   // K rows (5 KB)
  __shared__ float ldsW[TP * HDP];   // q_a-scaled V rows (5 KB)

  const int a    = blockIdx.x;     // 0..Tz-1
  const int i    = blockIdx.y;     // 0..BHz-1
  const int lane = threadIdx.x & 31;
  const int half = lane >> 4;
  const int l15  = lane & 15;

  const float* qa = qh + (i * TP + a) * HDP;
  const float* Kb = kh + i * TP * HDP;
  const float* Vb = vh + i * TP * HDP;

  // --- stage K tile into LDS ---
#ifdef ASYNC_LDS
  {
#pragma unroll
    for (int j = 0; j < (TP * HDP) / (32 * 4); ++j) {   // 10 x (32 lanes x 16B)
      const float* gsrc = Kb + j * 128 + lane * 4;
      float* ldst = &ldsK[j * 128 + lane * 4];
      __builtin_amdgcn_global_load_async_to_lds_b128(
          (__attribute__((address_space(1))) v4i*)gsrc,
          (__attribute__((address_space(3))) v4i*)ldst, 0, 0);
    }
  }
#else
  for (int c = lane; c < TP; c += 32) {
#pragma unroll
    for (int t = 0; t < HDP; ++t) ldsK[c * HDP + t] = Kb[c * HDP + t];
  }
#endif

  float qv[HDP];
#pragma unroll
  for (int t = 0; t < HDP; ++t) qv[t] = qa[t];

  // --- W[c][t] = q_a[t] * V[c][t] into LDS (overlaps async K copy) ---
  for (int c = lane; c < TP; c += 32) {
#pragma unroll
    for (int t = 0; t < HDP; ++t) ldsW[c * HDP + t] = qv[t] * Vb[c * HDP + t];
  }

#ifdef ASYNC_LDS
#if __has_builtin(__builtin_amdgcn_s_wait_asynccnt)
  __builtin_amdgcn_s_wait_asynccnt(0);
#else
  asm volatile("s_wait_asynccnt 0" ::: "memory");
#endif
#endif
  __syncthreads();   // single wave: memory-ordering fence (barrier lowers to s_nop)

  for (int bt = 0; bt < TP / 16; ++bt) {
    // A fragment (16x4 f32 layout): lane l15 = row M, VGPR0: K = kbase+2*half, VGPR1: +1
    const int arow = (bt * 16 + l15) * HDP;
    v2f af0, af1;
    af0[0] = ldsK[arow + 2 * half];     af0[1] = ldsK[arow + 2 * half + 1];
    af1[0] = ldsK[arow + 4 + 2 * half]; af1[1] = ldsK[arow + 5 + 2 * half];

    float sum[8], mx[8];
#pragma unroll
    for (int r = 0; r < 8; ++r) { sum[r] = 0.f; mx[r] = -1e30f; }

    for (int ct = 0; ct < TP / 16; ++ct) {
      const int c = ct * 16 + l15;           // column N = l15
      const int crow = c * HDP;
      // B fragment (4x16): VGPR0 row kbase+2*half, VGPR1 row kbase+2*half+1
      v2f bf0, bf1;
      bf0[0] = ldsW[crow + 2 * half];     bf0[1] = ldsW[crow + 2 * half + 1];
      bf1[0] = ldsW[crow + 4 + 2 * half]; bf1[1] = ldsW[crow + 5 + 2 * half];

      v8f acc = {};
      acc = __builtin_amdgcn_wmma_f32_16x16x4_f32(false, af0, false, bf0,
                                                  (short)0, acc, false, false);
      acc = __builtin_amdgcn_wmma_f32_16x16x4_f32(false, af1, false, bf1,
                                                  (short)0, acc, false, false);

      const bool cv = (c < Tz);
#pragma unroll
      for (int r = 0; r < 8; ++r) {
        float val = acc[r];
        sum[r] += cv ? val : 0.f;
        mx[r]   = fmaxf(mx[r], cv ? val : -1e30f);
      }
    }

    // reduce over the 16 lanes of each half (columns of this tile row block)
#pragma unroll
    for (int r = 0; r < 8; ++r) {
      float s0 = sum[r], m0 = mx[r];
      s0 += __shfl_xor(s0, 1);  m0 = fmaxf(m0, __shfl_xor(m0, 1));
      s0 += __shfl_xor(s0, 2);  m0 = fmaxf(m0, __shfl_xor(m0, 2));
      s0 += __shfl_xor(s0, 4);  m0 = fmaxf(m0, __shfl_xor(m0, 4));
      s0 += __shfl_xor(s0, 8);  m0 = fmaxf(m0, __shfl_xor(m0, 8));
      sum[r] = s0; mx[r] = m0;
    }
    if (l15 == 0) {
#pragma unroll
      for (int r = 0; r < 8; ++r) {
        int brow = bt * 16 + r + 8 * half;   // C layout: VGPR r = rows r, r+8
        if (brow < Tz)
          fused[(i * Tz + a) * TP + brow] = sum[r] * (1.0f / Tz) + mx[r];
      }
    }
  }
}

// ---------------- row softmax over b (len Tz, stride-TP rows)
__global__ void softmax_kernel(float* __restrict__ fused) {
  int r = blockIdx.x * blockDim.x + threadIdx.x;
  if (r >= BHz * Tz) return;
  float* row = fused + r * TP;
  float mx = -1e30f;
  for (int b = 0; b < Tz; ++b) mx = fmaxf(mx, row[b]);
  float s = 0.f;
  for (int b = 0; b < Tz; ++b) { float e = __expf(row[b] - mx); row[b] = e; s += e; }
  float inv = 1.f / s;
  for (int b = 0; b < Tz; ++b) row[b] *= inv;
}

// ---------------- attn = fused @ qh, merged back to [T,B,D]
__global__ void attn_out_kernel(const float* __restrict__ fused, const float* __restrict__ qh,
                                float* __restrict__ outc) {
  int idx = blockIdx.x * blockDim.x + threadIdx.x;
  if (idx >= BHz * Tz * HDz) return;
  int e = idx % HDz;
  int a = (idx / HDz) % Tz;
  int i = idx / (HDz * Tz);
  const float* row = fused + (i * Tz + a) * TP;
  const float* qb  = qh + i * TP * HDP + e;
  __builtin_prefetch(row, 0, 0);          // -> global_prefetch_b8
  __builtin_prefetch(qb, 0, 0);
  float s = 0.f;
  for (int b = 0; b < Tz; ++b) s += row[b] * qb[b * HDP];
  int bb = i / Hz, hh = i % Hz;
  outc[(a * Bz + bb) * Dz + hh * HDz + e] = s;
}

// ---------------- x1 = res + attn @ ow^T + ob
__global__ void resproj_kernel(const float* __restrict__ h, const float* __restrict__ attnc,
                               const float* __restrict__ ow, const float* __restrict__ ob,
                               float* __restrict__ x1) {
  int idx = blockIdx.x * blockDim.x + threadIdx.x;
  if (idx >= NTBD) return;
  int d = idx % Dz;
  int r = idx / Dz;
  const float* ar = attnc + r * Dz;
  const float* wr = ow + d * Dz;
  float s = ob[d];
  for (int e = 0; e < Dz; ++e) s += ar[e] * wr[e];
  x1[idx] = h[idx] + s;
}

// ---------------- FFN layer 1: relu(xn @ w1^T + b1), [rows,120]
__global__ void ffn1_kernel(const float* __restrict__ xn, const float* __restrict__ w1,
                            const float* __restrict__ b1, float* __restrict__ t1) {
  int idx = blockIdx.x * blockDim.x + threadIdx.x;
  if (idx >= ROWS * 4 * Dz) return;
  int j = idx % (4 * Dz);
  int r = idx / (4 * Dz);
  const float* xr = xn + r * Dz;
  const float* wr = w1 + j * Dz;
  float s = b1[j];
  for (int e = 0; e < Dz; ++e) s += xr[e] * wr[e];
  t1[idx] = fmaxf(s, 0.f);
}

// ---------------- FFN layer 2 + residual -> new h
__global__ void ffn2_kernel(const float* __restrict__ x1, const float* __restrict__ t1,
                            const float* __restrict__ w2, const float* __restrict__ b2,
                            float* __restrict__ hout) {
  int idx = blockIdx.x * blockDim.x + threadIdx.x;
  if (idx >= NTBD) return;
  int d = idx % Dz;
  int r = idx / Dz;
  const float* tr = t1 + r * 4 * Dz;
  const float* wr = w2 + d * 4 * Dz;
  float s = b2[d];
  for (int j = 0; j < 4 * Dz; ++j) s += tr[j] * wr[j];
  hout[idx] = x1[idx] + s;
}

// ---------------- grab last timestep into last_hs[b][s*D + d]
__global__ void lasths_kernel(const float* __restrict__ h, float* __restrict__ last, int s) {
  int idx = blockIdx.x * blockDim.x + threadIdx.x;
  if (idx >= Bz * Dz) return;
  int d = idx % Dz;
  int b = idx / Dz;
  last[b * (2 * Dz) + s * Dz + d] = h[((Tz - 1) * Bz + b) * Dz + d];
}

// ---------------- final head: proj = relu(hs @ p1^T + b1) @ p2^T + b2 + hs; out = proj @ ow^T + ob
__global__ void final_kernel(const float* __restrict__ last, const float* __restrict__ p1w,
                             const float* __restrict__ p1b, const float* __restrict__ p2w,
                             const float* __restrict__ p2b, const float* __restrict__ oww,
                             const float* __restrict__ obb, float* __restrict__ dout) {
  int b = blockIdx.x * blockDim.x + threadIdx.x;
  if (b >= Bz) return;
  const float* lr = last + b * 2 * Dz;
  float t0[2 * Dz];
  for (int j = 0; j < 2 * Dz; ++j) {
    float s = p1b[j];
    for (int e = 0; e < 2 * Dz; ++e) s += lr[e] * p1w[j * 2 * Dz + e];
    t0[j] = fmaxf(s, 0.f);
  }
  float o = obb[0];
  for (int j = 0; j < 2 * Dz; ++j) {
    float s = p2b[j];
    for (int e = 0; e < 2 * Dz; ++e) s += t0[e] * p2w[j * 2 * Dz + e];
    s += lr[j];
    dout[Bz + b * 2 * Dz + j] = s;   // last_hs output (after the 8 'out' floats)
    o += s * oww[j];
  }
  dout[b] = o;                        // out[b][0]
}

// ---- workspace layout (float offsets) ----
#define OFF_PL 0
#define OFF_PA 36000
#define OFF_PV 72000
#define OFF_H  108000
#define OFF_X1 144000
#define OFF_XN 180000
#define OFF_KN 216000
#define OFF_VN 252000
#define OFF_Q  288000
#define OFF_K  324000
#define OFF_V  360000
#define OFF_AT 396000
#define OFF_QH 432000          // 40*160*8 = 51200
#define OFF_KH 483200
#define OFF_VH 534400
#define OFF_T1 585600          // 144000
#define OFF_FU 729600          // 40*150*160 = 960000

static inline int cdiv(int a, int b) { return (a + b - 1) / b; }

extern "C" void kernel_launch(void* const* d_in, const int* in_sizes, int n_in,
                              void* d_out, int out_size, void* d_ws, size_t ws_size,
                              hipStream_t stream) {
  const float* x_l   = (const float*)d_in[0];
  const float* x_a   = (const float*)d_in[1];
  const float* x_v   = (const float*)d_in[2];
  const float* Wl    = (const float*)d_in[3];
  const float* Wa    = (const float*)d_in[4];
  const float* Wv    = (const float*)d_in[5];
  const float* in_w  = (const float*)d_in[6];
  const float* in_b  = (const float*)d_in[7];
  const float* out_w = (const float*)d_in[8];
  const float* out_b = (const float*)d_in[9];
  const float* l1w   = (const float*)d_in[10];
  const float* l1b   = (const float*)d_in[11];
  const float* l2w   = (const float*)d_in[12];
  const float* l2b   = (const float*)d_in[13];
  const float* n1g   = (const float*)d_in[14];
  const float* n1b   = (const float*)d_in[15];
  const float* n2g   = (const float*)d_in[16];
  const float* n2b   = (const float*)d_in[17];
  const float* p1w   = (const float*)d_in[18];
  const float* p1b   = (const float*)d_in[19];
  const float* p2w   = (const float*)d_in[20];
  const float* p2b   = (const float*)d_in[21];
  const float* oww   = (const float*)d_in[22];
  const float* obb   = (const float*)d_in[23];
  float* dout = (float*)d_out;

  float* ws = (float*)d_ws;
  float* pl = ws + OFF_PL; float* pa = ws + OFF_PA; float* pv = ws + OFF_PV;
  float* h  = ws + OFF_H;  float* x1 = ws + OFF_X1;
  float* xn = ws + OFF_XN; float* kn = ws + OFF_KN; float* vn = ws + OFF_VN;
  float* q  = ws + OFF_Q;  float* k  = ws + OFF_K;  float* v  = ws + OFF_V;
  float* at = ws + OFF_AT;
  float* qh = ws + OFF_QH; float* kh = ws + OFF_KH; float* vh = ws + OFF_VH;
  float* t1 = ws + OFF_T1; float* fu = ws + OFF_FU;
  float* last = fu + 960000;   // 480 floats

  const int TB = 256;
  const float scale = rsqrtf((float)HDz);

  proj_kernel<<<cdiv(NTBD, TB), TB, 0, stream>>>(x_l, Wl, pl, 300);
  proj_kernel<<<cdiv(NTBD, TB), TB, 0, stream>>>(x_a, Wa, pa, 74);
  proj_kernel<<<cdiv(NTBD, TB), TB, 0, stream>>>(x_v, Wv, pv, 35);

  for (int s = 0; s < 2; ++s) {
    const float* kk = (s == 0) ? pa : pv;
    const float* vv = (s == 0) ? pv : pa;
    (void)hipMemcpyAsync(h, pl, NTBD * sizeof(float), hipMemcpyDeviceToDevice, stream);

    for (int l = 0; l < Lz; ++l) {
      int sl = s * Lz + l;
      const float* iw = in_w + sl * 3 * Dz * Dz;
      const float* ib = in_b + sl * 3 * Dz;
      const float* ow = out_w + sl * Dz * Dz;
      const float* ob = out_b + sl * Dz;
      const float* w1 = l1w + sl * 4 * Dz * Dz;
      const float* b1 = l1b + sl * 4 * Dz;
      const float* w2 = l2w + sl * Dz * 4 * Dz;
      const float* b2 = l2b + sl * Dz;
      const float* g1 = n1g + sl * Dz; const float* bb1 = n1b + sl * Dz;
      const float* g2 = n2g + sl * Dz; const float* bb2 = n2b + sl * Dz;

      ln_kernel<<<cdiv(ROWS, TB), TB, 0, stream>>>(h,  g1, bb1, xn);
      ln_kernel<<<cdiv(ROWS, TB), TB, 0, stream>>>(kk, g1, bb1, kn);
      ln_kernel<<<cdiv(ROWS, TB), TB, 0, stream>>>(vv, g1, bb1, vn);
      qkv_kernel<<<cdiv(ROWS * Dz, TB), TB, 0, stream>>>(xn, kn, vn, iw, ib, q, k, v, scale);
      heads_kernel<<<cdiv(BHz * TP * HDP, TB), TB, 0, stream>>>(q, k, v, qh, kh, vh);
      attn_scores_kernel<<<dim3(Tz, BHz), 32, 0, stream>>>(qh, kh, vh, fu);
      softmax_kernel<<<cdiv(BHz * Tz, TB), TB, 0, stream>>>(fu);
      attn_out_kernel<<<cdiv(BHz * Tz * HDz, TB), TB, 0, stream>>>(fu, qh, at);
      resproj_kernel<<<cdiv(NTBD, TB), TB, 0, stream>>>(h, at, ow, ob, x1);
      ln_kernel<<<cdiv(ROWS, TB), TB, 0, stream>>>(x1, g2, bb2, xn);
      ffn1_kernel<<<cdiv(ROWS * 4 * Dz, TB), TB, 0, stream>>>(xn, w1, b1, t1);
      ffn2_kernel<<<cdiv(NTBD, TB), TB, 0, stream>>>(x1, t1, w2, b2, h);
    }
    lasths_kernel<<<cdiv(Bz * Dz, TB), TB, 0, stream>>>(h, last, s);
  }

  final_kernel<<<1, 32, 0, stream>>>(last, p1w, p1b, p2w, p2b, oww, obb, dout);
}